// GraphSAGE2_80676665688553
// MI455X (gfx1250) — compile-verified
//
#include <hip/hip_runtime.h>

typedef __attribute__((ext_vector_type(2))) float v2f;
typedef __attribute__((ext_vector_type(8))) float v8f;

#define NN 100000
#define NE 1600000
#define DD 64
#define DOUT 40
#define NTILES (NN / 16)   // 6250, exact

// ---------------- zero workspace ----------------
__global__ void zero_kernel(float* p, long n) {
    long i = (long)blockIdx.x * blockDim.x + threadIdx.x;
    long stride = (long)gridDim.x * blockDim.x;
    for (; i < n; i += stride) p[i] = 0.0f;
}

// ---------------- degree count ----------------
__global__ void count_kernel(const long long* dst, float* cnt, int ne) {
    int e = blockIdx.x * blockDim.x + threadIdx.x;
    if (e < ne) atomicAdd(&cnt[(int)dst[e]], 1.0f);
}

// ---------------- scatter-add feat[src] -> agg[dst] ----------------
// thread t handles edge t>>4, features [(t&15)*4, +4)
__global__ void scatter_kernel(const float* __restrict__ feat,
                               const long long* __restrict__ src,
                               const long long* __restrict__ dst,
                               float* agg, int ne) {
    int t = blockIdx.x * blockDim.x + threadIdx.x;
    int e = t >> 4;
    if (e >= ne) return;
    int f = (t & 15) * 4;
    int s = (int)src[e];
    int d = (int)dst[e];
    const float4 v = *(const float4*)(feat + (long)s * DD + f);
    float* o = agg + (long)d * DD + f;
    atomicAdd(o + 0, v.x);
    atomicAdd(o + 1, v.y);
    atomicAdd(o + 2, v.z);
    atomicAdd(o + 3, v.w);
}

// ---------------- layer1 GEMM: h = (agg/cnt)@W1l^T + x@W1r^T + b1, + BN stats ----------------
// one wave32 per 16-row tile; V_WMMA_F32_16X16X4_F32, 16 k-steps per matmul
__global__ void gemm1_kernel(const float* __restrict__ agg, const float* __restrict__ x,
                             const float* __restrict__ Wl, const float* __restrict__ Wr,
                             const float* __restrict__ bias,
                             const float* __restrict__ cnt,
                             float* h, float* stats, int ntiles) {
    int wave = (blockIdx.x * blockDim.x + threadIdx.x) >> 5;
    int lane = threadIdx.x & 31;
    if (wave >= ntiles) return;

    int m0 = wave * 16;
    int row = m0 + (lane & 15);        // A: lanes 0-15 and 16-31 both map M=lane&15
    int khalf = (lane >> 4) * 2;       // lanes 0-15: K=0,1 ; lanes 16-31: K=2,3
    float rc = 1.0f / fmaxf(cnt[row], 1.0f);

    const float* arow = agg + (long)row * DD + khalf;
    const float* xrow = x   + (long)row * DD + khalf;
    v2f afrag[16], xfrag[16];
#pragma unroll
    for (int k = 0; k < 16; k++) {
        v2f a = *(const v2f*)(arow + k * 4);
        afrag[k] = a * rc;                      // fold mean-division into A
        xfrag[k] = *(const v2f*)(xrow + k * 4);
    }

    int ncol = lane & 15;
    int mbase = m0 + ((lane >> 4) ? 8 : 0);     // C VGPR r -> row mbase+r
#pragma unroll
    for (int nt = 0; nt < 4; nt++) {
        int n = nt * 16 + ncol;
        const float* wl = Wl + (long)n * DD + khalf;   // B[k][n] = W[n*64+k]
        const float* wr = Wr + (long)n * DD + khalf;
        v8f c = {};
#pragma unroll
        for (int k = 0; k < 16; k++) {
            v2f bl = *(const v2f*)(wl + k * 4);
            c = __builtin_amdgcn_wmma_f32_16x16x4_f32(false, afrag[k], false, bl,
                                                      (short)0, c, false, false);
        }
#pragma unroll
        for (int k = 0; k < 16; k++) {
            v2f br = *(const v2f*)(wr + k * 4);
            c = __builtin_amdgcn_wmma_f32_16x16x4_f32(false, xfrag[k], false, br,
                                                      (short)0, c, false, false);
        }
        float b = bias[n];
        float s = 0.0f, s2 = 0.0f;
#pragma unroll
        for (int r = 0; r < 8; r++) {
            float v = c[r] + b;
            h[(long)(mbase + r) * DD + n] = v;
            s += v;
            s2 += v * v;
        }
        atomicAdd(&stats[n], s);
        atomicAdd(&stats[DD + n], s2);
    }
}

// ---------------- BatchNorm (batch stats) + ReLU, in place ----------------
__global__ void bn_kernel(float* h, const float* __restrict__ stats,
                          const float* __restrict__ gamma, const float* __restrict__ beta,
                          long total) {
    long i = (long)blockIdx.x * blockDim.x + threadIdx.x;
    if (i >= total) return;
    int c = (int)(i & (DD - 1));
    const float invN = 1.0f / (float)NN;
    float mu = stats[c] * invN;
    float var = stats[DD + c] * invN - mu * mu;
    float rs = rsqrtf(var + 1e-5f);
    float v = (h[i] - mu) * rs * gamma[c] + beta[c];
    h[i] = fmaxf(v, 0.0f);
}

// ---------------- layer2 GEMM: out = (agg2/cnt)@W2l^T + h@W2r^T + b2  (N x 40) ----------------
__global__ void gemm2_kernel(const float* __restrict__ agg, const float* __restrict__ hbn,
                             const float* __restrict__ Wl, const float* __restrict__ Wr,
                             const float* __restrict__ bias,
                             const float* __restrict__ cnt,
                             float* out, int ntiles) {
    int wave = (blockIdx.x * blockDim.x + threadIdx.x) >> 5;
    int lane = threadIdx.x & 31;
    if (wave >= ntiles) return;

    int m0 = wave * 16;
    int row = m0 + (lane & 15);
    int khalf = (lane >> 4) * 2;
    float rc = 1.0f / fmaxf(cnt[row], 1.0f);

    const float* arow = agg + (long)row * DD + khalf;
    const float* xrow = hbn + (long)row * DD + khalf;
    v2f afrag[16], xfrag[16];
#pragma unroll
    for (int k = 0; k < 16; k++) {
        v2f a = *(const v2f*)(arow + k * 4);
        afrag[k] = a * rc;
        xfrag[k] = *(const v2f*)(xrow + k * 4);
    }

    int ncol = lane & 15;
    int mbase = m0 + ((lane >> 4) ? 8 : 0);
#pragma unroll
    for (int nt = 0; nt < 3; nt++) {           // 48 cols, mask at 40
        int n = nt * 16 + ncol;
        bool valid = (n < DOUT);
        const float* wl = Wl + (long)(valid ? n : 0) * DD + khalf;
        const float* wr = Wr + (long)(valid ? n : 0) * DD + khalf;
        v8f c = {};
#pragma unroll
        for (int k = 0; k < 16; k++) {
            v2f bl = valid ? *(const v2f*)(wl + k * 4) : (v2f){0.0f, 0.0f};
            c = __builtin_amdgcn_wmma_f32_16x16x4_f32(false, afrag[k], false, bl,
                                                      (short)0, c, false, false);
        }
#pragma unroll
        for (int k = 0; k < 16; k++) {
            v2f br = valid ? *(const v2f*)(wr + k * 4) : (v2f){0.0f, 0.0f};
            c = __builtin_amdgcn_wmma_f32_16x16x4_f32(false, xfrag[k], false, br,
                                                      (short)0, c, false, false);
        }
        if (valid) {
            float b = bias[n];
#pragma unroll
            for (int r = 0; r < 8; r++)
                out[(long)(mbase + r) * DOUT + n] = c[r] + b;
        }
    }
}

extern "C" void kernel_launch(void* const* d_in, const int* in_sizes, int n_in,
                              void* d_out, int out_size, void* d_ws, size_t ws_size,
                              hipStream_t stream) {
    const float*     x     = (const float*)d_in[0];
    const long long* ei    = (const long long*)d_in[1];   // int64 [2, E]
    const float*     W1l   = (const float*)d_in[2];
    const float*     W1r   = (const float*)d_in[3];
    const float*     b1    = (const float*)d_in[4];
    const float*     gamma = (const float*)d_in[5];
    const float*     beta  = (const float*)d_in[6];
    const float*     W2l   = (const float*)d_in[7];
    const float*     W2r   = (const float*)d_in[8];
    const float*     b2    = (const float*)d_in[9];

    const long long* src = ei;
    const long long* dst = ei + NE;

    float* ws    = (float*)d_ws;
    float* buf1  = ws;                               // agg1 -> h -> h_bn (in place), N*64
    float* buf2  = ws + (size_t)NN * DD;             // agg2, N*64
    float* cnt   = ws + (size_t)2 * NN * DD;         // N
    float* stats = cnt + NN;                         // 128 (sum, sumsq)

    const long nzero = (long)2 * NN * DD + NN + 2 * DD;
    zero_kernel<<<2048, 256, 0, stream>>>(ws, nzero);

    count_kernel<<<(NE + 255) / 256, 256, 0, stream>>>(dst, cnt, NE);

    // layer 1
    scatter_kernel<<<(NE * 16 + 255) / 256, 256, 0, stream>>>(x, src, dst, buf1, NE);
    {
        int waves_per_block = 256 / 32;
        int grid = (NTILES + waves_per_block - 1) / waves_per_block;
        gemm1_kernel<<<grid, 256, 0, stream>>>(buf1, x, W1l, W1r, b1, cnt, buf1, stats, NTILES);
    }
    bn_kernel<<<((long)NN * DD + 255) / 256, 256, 0, stream>>>(buf1, stats, gamma, beta, (long)NN * DD);

    // layer 2
    scatter_kernel<<<(NE * 16 + 255) / 256, 256, 0, stream>>>(buf1, src, dst, buf2, NE);
    {
        int waves_per_block = 256 / 32;
        int grid = (NTILES + waves_per_block - 1) / waves_per_block;
        gemm2_kernel<<<grid, 256, 0, stream>>>(buf2, buf1, W2l, W2r, b2, cnt, (float*)d_out, NTILES);
    }
}